// PointSetAttention_34471407517998
// MI455X (gfx1250) — compile-verified
//
#include <hip/hip_runtime.h>
#include <hip/hip_bf16.h>
#include <limits.h>

// ---------------- problem constants (from reference) ----------------
#define N_NODES 50000
#define N_EDGES 1600000
#define FDIM    128
#define NH      8           // heads
#define NP      4           // point dim
#define EDIM    32
#define HP      32          // NH*NP
#define NROWS   (N_NODES*4) // 200000 rows of the [N,4,*] tensors
#define NTILES  (NROWS/16)  // 12500, exact
#define INV_DS        0.1f
#define SCALAR_SCALE  0.35355339059327373f   // (2*4)^-0.5
#define POINT_SCALE   0.16666666666666666f   // (2*4*4.5)^-0.5 = 1/6

typedef __attribute__((ext_vector_type(2))) float v2f;
typedef __attribute__((ext_vector_type(8))) float v8f;

// CDNA5 fp32 WMMA: D(16x16,f32) = A(16x4,f32) x B(4x16,f32) + C
// 8-arg pattern: (neg_a, A, neg_b, B, c_mod, C, reuse_a, reuse_b)
static __device__ inline v8f wmma_f32(v2f a, v2f b, v8f c) {
  return __builtin_amdgcn_wmma_f32_16x16x4_f32(false, a, false, b, (short)0, c,
                                               false, false);
}

// monotone float<->int ordering for atomicMax-based segment max
static __device__ inline int f2ord(float f) {
  int i = __float_as_int(f);
  return i >= 0 ? i : (i ^ 0x7fffffff);
}
static __device__ inline float ord2f(int o) {
  int i = o >= 0 ? o : (o ^ 0x7fffffff);
  return __int_as_float(i);
}

// ---------------- kernel 0: init scratch ----------------
__global__ __launch_bounds__(256) void init_kernel(float* __restrict__ res,
                                                   int* __restrict__ mord,
                                                   float* __restrict__ denom) {
  long t = (long)blockIdx.x * 256 + threadIdx.x;
  if (t < (long)NROWS * HP) res[t] = 0.0f;
  if (t < (long)N_NODES * NH) { mord[t] = INT_MIN; denom[t] = 0.0f; }
}

// ---------------- kernel 1: q/k/v projections (WMMA f32) ----------------
// X[200000x128] @ W[128x32]; one wave handles a 16-row tile, both 16-col
// halves of the 32-wide output, for all three weight matrices.
__global__ __launch_bounds__(256) void qkv_proj_kernel(
    const float* __restrict__ x_q, const float* __restrict__ x_k,
    const float* __restrict__ Wq, const float* __restrict__ Wk,
    const float* __restrict__ Wv,
    const float* __restrict__ pcq, const float* __restrict__ pck,
    float* __restrict__ qo, float* __restrict__ ko, float* __restrict__ vo) {
  // weights transposed in LDS: sW[m][col*128 + row] = W[row*32 + col]
  __shared__ float sW[3][FDIM * HP];  // 48 KB
  for (int i = threadIdx.x; i < FDIM * HP; i += 256) {
    int row = i >> 5, col = i & 31;
    sW[0][col * FDIM + row] = Wq[i];
    sW[1][col * FDIM + row] = Wk[i];
    sW[2][col * FDIM + row] = Wv[i];
  }
  __syncthreads();

  int wave = threadIdx.x >> 5;
  int lane = threadIdx.x & 31;
  int half = lane >> 4;     // lane group 0..15 vs 16..31
  int l15  = lane & 15;
  int tile = blockIdx.x * 8 + wave;
  if (tile >= NTILES) return;
  int row0 = tile * 16;
  int arow = row0 + l15;    // A fragment row for this lane

  const float* aq = x_q + (size_t)arow * FDIM + 2 * half;
  const float* ak = x_k + (size_t)arow * FDIM + 2 * half;
  const float* bq0 = &sW[0][(l15     ) * FDIM + 2 * half];
  const float* bq1 = &sW[0][(l15 + 16) * FDIM + 2 * half];
  const float* bk0 = &sW[1][(l15     ) * FDIM + 2 * half];
  const float* bk1 = &sW[1][(l15 + 16) * FDIM + 2 * half];
  const float* bv0 = &sW[2][(l15     ) * FDIM + 2 * half];
  const float* bv1 = &sW[2][(l15 + 16) * FDIM + 2 * half];

  v8f cq0 = {}, cq1 = {}, ck0 = {}, ck1 = {}, cv0 = {}, cv1 = {};
#pragma unroll 4
  for (int s = 0; s < 32; ++s) {           // K = 4*s .. 4*s+3
    v2f a_q = *(const v2f*)(aq + 4 * s);
    v2f a_k = *(const v2f*)(ak + 4 * s);
    cq0 = wmma_f32(a_q, *(const v2f*)(bq0 + 4 * s), cq0);
    cq1 = wmma_f32(a_q, *(const v2f*)(bq1 + 4 * s), cq1);
    ck0 = wmma_f32(a_k, *(const v2f*)(bk0 + 4 * s), ck0);
    ck1 = wmma_f32(a_k, *(const v2f*)(bk1 + 4 * s), ck1);
    cv0 = wmma_f32(a_k, *(const v2f*)(bv0 + 4 * s), cv0);
    cv1 = wmma_f32(a_k, *(const v2f*)(bv1 + 4 * s), cv1);
  }

  // epilogue: add point centers / DS to rows c = 1..3 (scalar row c=0 skips)
#pragma unroll
  for (int i = 0; i < 8; ++i) {
    int r = row0 + i + 8 * half;           // D row per ISA C/D layout
    int c = r & 3;
    int node = r >> 2;
    float ctrq = c ? pcq[node * 3 + (c - 1)] * INV_DS : 0.0f;
    float ctrk = c ? pck[node * 3 + (c - 1)] * INV_DS : 0.0f;
    size_t base = (size_t)r * HP;
    qo[base + l15]      = cq0[i] + ctrq;
    qo[base + 16 + l15] = cq1[i] + ctrq;
    ko[base + l15]      = ck0[i] + ctrk;
    ko[base + 16 + l15] = ck1[i] + ctrk;
    vo[base + l15]      = cv0[i] + ctrk;
    vo[base + 16 + l15] = cv1[i] + ctrk;
  }
}

// ---------------- kernel 2a: per-edge logits + segment max ----------------
__global__ __launch_bounds__(256) void edge_logits_kernel(
    const float* __restrict__ q, const float* __restrict__ k,
    const float* __restrict__ x_edge, const float* __restrict__ We,
    const float* __restrict__ point_weights, const int* __restrict__ ei,
    float* __restrict__ lg, int* __restrict__ mord) {
  __shared__ float sWe[EDIM * NH];
  __shared__ float sPw[NH];
  if (threadIdx.x < EDIM * NH) sWe[threadIdx.x] = We[threadIdx.x];
  if (threadIdx.x < NH) {
    float w = point_weights[threadIdx.x];
    sPw[threadIdx.x] = (w > 20.0f) ? w : log1pf(expf(w));  // softplus
  }
  __syncthreads();

  int e = blockIdx.x * 256 + threadIdx.x;
  if (e >= N_EDGES) return;
  int src = ei[e];
  int dst = ei[N_EDGES + e];
  const float* qr = q + (size_t)dst * 128;   // [4][32] rows of node dst
  const float* kr = k + (size_t)src * 128;
  const float* xe = x_edge + (size_t)e * EDIM;

  float lgv[NH];
#pragma unroll
  for (int h = 0; h < NH; ++h) {
    float sdot = 0.0f;
#pragma unroll
    for (int p = 0; p < NP; ++p) sdot += qr[h * NP + p] * kr[h * NP + p];
    float pl = 0.0f;
#pragma unroll
    for (int c = 1; c <= 3; ++c)
#pragma unroll
      for (int p = 0; p < NP; ++p) {
        float d = qr[c * 32 + h * NP + p] - kr[c * 32 + h * NP + p];
        pl += d * d;
      }
    lgv[h] = sdot * SCALAR_SCALE - 0.5f * pl * POINT_SCALE * sPw[h];
  }
#pragma unroll 8
  for (int j = 0; j < EDIM; ++j) {
    float xv = xe[j];
#pragma unroll
    for (int h = 0; h < NH; ++h) lgv[h] += xv * sWe[j * NH + h];
  }
#pragma unroll
  for (int h = 0; h < NH; ++h) {
    lg[(size_t)e * NH + h] = lgv[h];
    atomicMax(&mord[dst * NH + h], f2ord(lgv[h]));
  }
}

// ---------------- kernel 2b: exp and segment sum ----------------
__global__ __launch_bounds__(256) void edge_exp_kernel(
    const int* __restrict__ ei, const int* __restrict__ mord,
    float* __restrict__ lg, float* __restrict__ denom) {
  long t = (long)blockIdx.x * 256 + threadIdx.x;  // over E*NH
  if (t >= (long)N_EDGES * NH) return;
  int e = (int)(t >> 3);
  int h = (int)(t & 7);
  int dst = ei[N_EDGES + e];
  float m = ord2f(mord[dst * NH + h]);
  float ex = expf(lg[t] - m);
  lg[t] = ex;
  atomicAdd(&denom[dst * NH + h], ex);
}

// ---------------- kernel 2c: weighted scatter-aggregate ----------------
__global__ __launch_bounds__(256) void edge_aggr_kernel(
    const int* __restrict__ ei, const float* __restrict__ lg,
    const float* __restrict__ denom, const float* __restrict__ v,
    float* __restrict__ res) {
  long t = (long)blockIdx.x * 256 + threadIdx.x;  // over E*32
  if (t >= (long)N_EDGES * HP) return;
  int e = (int)(t >> 5);
  int d = (int)(t & 31);
  int h = d >> 2;
  int src = ei[e];
  int dst = ei[N_EDGES + e];
  float alpha = lg[(size_t)e * NH + h] / denom[dst * NH + h];
  const float* vr = v + (size_t)src * 128;
  float* rr = res + (size_t)dst * 128;
#pragma unroll
  for (int c = 0; c < 4; ++c)
    atomicAdd(&rr[c * 32 + d], alpha * vr[c * 32 + d]);
}

// ---------------- kernel 3: output projection (WMMA f32) ----------------
// (res - centers)[200000x32] @ Wo[32x128]
__global__ __launch_bounds__(256) void out_proj_kernel(
    const float* __restrict__ res, const float* __restrict__ pcq,
    const float* __restrict__ Wo, float* __restrict__ out) {
  __shared__ float sWo[FDIM * HP];  // transposed: sWo[col*32 + row], 16 KB
  for (int i = threadIdx.x; i < HP * FDIM; i += 256) {
    int row = i >> 7, col = i & 127;    // Wo is [32][128]
    sWo[col * HP + row] = Wo[i];
  }
  __syncthreads();

  int wave = threadIdx.x >> 5;
  int lane = threadIdx.x & 31;
  int half = lane >> 4;
  int l15  = lane & 15;
  int tile = blockIdx.x * 8 + wave;
  if (tile >= NTILES) return;
  int row0 = tile * 16;
  int ar = row0 + l15;
  int c = ar & 3;
  int node = ar >> 2;
  float ctr = c ? pcq[node * 3 + (c - 1)] * INV_DS : 0.0f;  // subtract centers_q/DS

  // preload the 8 A fragments (K = 0..31)
  v2f afr[8];
#pragma unroll
  for (int s = 0; s < 8; ++s) {
    v2f a = *(const v2f*)(res + (size_t)ar * HP + 4 * s + 2 * half);
    a.x -= ctr;
    a.y -= ctr;
    afr[s] = a;
  }

#pragma unroll
  for (int nt = 0; nt < 8; ++nt) {   // 8 column tiles of 16 -> 128 outputs
    v8f acc = {};
#pragma unroll
    for (int s = 0; s < 8; ++s) {
      v2f b = *(const v2f*)(&sWo[(nt * 16 + l15) * HP + 4 * s + 2 * half]);
      acc = wmma_f32(afr[s], b, acc);
    }
#pragma unroll
    for (int i = 0; i < 8; ++i) {
      int r = row0 + i + 8 * half;
      out[(size_t)r * FDIM + nt * 16 + l15] = acc[i];
    }
  }
}

// ---------------- host launch ----------------
extern "C" void kernel_launch(void* const* d_in, const int* in_sizes, int n_in,
                              void* d_out, int out_size, void* d_ws,
                              size_t ws_size, hipStream_t stream) {
  const float* x_k  = (const float*)d_in[0];
  const float* x_q  = (const float*)d_in[1];
  const float* pck  = (const float*)d_in[2];
  const float* pcq  = (const float*)d_in[3];
  const float* x_e  = (const float*)d_in[4];
  const int*   ei   = (const int*)d_in[5];
  const float* Wq   = (const float*)d_in[6];
  const float* Wk   = (const float*)d_in[7];
  const float* Wv   = (const float*)d_in[8];
  const float* We   = (const float*)d_in[9];
  const float* pw   = (const float*)d_in[10];
  const float* Wo   = (const float*)d_in[11];
  float* out = (float*)d_out;

  // workspace layout (~157 MB)
  float* q     = (float*)d_ws;
  float* k     = q + (size_t)NROWS * HP;
  float* v     = k + (size_t)NROWS * HP;
  float* res   = v + (size_t)NROWS * HP;
  float* lg    = res + (size_t)NROWS * HP;
  int*   mord  = (int*)(lg + (size_t)N_EDGES * NH);
  float* denom = (float*)(mord + (size_t)N_NODES * NH);

  {
    long n = (long)NROWS * HP;  // covers res; mord/denom are smaller
    init_kernel<<<(int)((n + 255) / 256), 256, 0, stream>>>(res, mord, denom);
  }
  qkv_proj_kernel<<<(NTILES + 7) / 8, 256, 0, stream>>>(
      x_q, x_k, Wq, Wk, Wv, pcq, pck, q, k, v);
  edge_logits_kernel<<<(N_EDGES + 255) / 256, 256, 0, stream>>>(
      q, k, x_e, We, pw, ei, lg, mord);
  {
    long n = (long)N_EDGES * NH;
    edge_exp_kernel<<<(int)((n + 255) / 256), 256, 0, stream>>>(ei, mord, lg,
                                                                denom);
  }
  {
    long n = (long)N_EDGES * HP;
    edge_aggr_kernel<<<(int)((n + 255) / 256), 256, 0, stream>>>(ei, lg, denom,
                                                                 v, res);
  }
  out_proj_kernel<<<(NTILES + 7) / 8, 256, 0, stream>>>(res, pcq, Wo, out);
}